// MMHCL_55430847922201
// MI455X (gfx1250) — compile-verified
//
#include <hip/hip_runtime.h>
#include <hip/hip_bf16.h>

#define DIM 64

typedef __attribute__((ext_vector_type(2))) float v2f;
typedef __attribute__((ext_vector_type(8))) float v8f;

// ---------------------------------------------------------------------------
// Guaranteed CDNA5 hardware FP32 atomic add (no-return form, device scope).
// Avoids any possibility of a CAS-loop expansion of atomicAdd(float*).
// Two adjacent dims share one 64-bit address via offset:4.
// ---------------------------------------------------------------------------
__device__ __forceinline__ void atomic_add2_f32(float* p, float v0, float v1) {
  asm volatile(
      "global_atomic_add_f32 %0, %1, off scope:SCOPE_DEV\n\t"
      "global_atomic_add_f32 %0, %2, off offset:4 scope:SCOPE_DEV"
      :
      : "v"(p), "v"(v0), "v"(v1)
      : "memory");
}

// ---------------------------------------------------------------------------
// Zero a buffer (float4 vectorized).
// ---------------------------------------------------------------------------
__global__ __launch_bounds__(256) void zero_k(float4* __restrict__ p, long n4) {
  long i = (long)blockIdx.x * blockDim.x + threadIdx.x;
  if (i < n4) {
    float4 z; z.x = 0.f; z.y = 0.f; z.z = 0.f; z.w = 0.f;
    p[i] = z;
  }
}

// ---------------------------------------------------------------------------
// ego0 = concat(user_ui, item_ui); acc = ego0   (float4 vectorized)
// ---------------------------------------------------------------------------
__global__ __launch_bounds__(256) void concat_k(const float4* __restrict__ u,
                                                const float4* __restrict__ it,
                                                float4* __restrict__ ego,
                                                float4* __restrict__ acc,
                                                long nu4, long n4) {
  long i = (long)blockIdx.x * blockDim.x + threadIdx.x;
  if (i < n4) {
    float4 v = (i < nu4) ? u[i] : it[i - nu4];
    ego[i] = v;
    acc[i] = v;
  }
}

// ---------------------------------------------------------------------------
// acc += e   (float4 vectorized)
// ---------------------------------------------------------------------------
__global__ __launch_bounds__(256) void add_k(float4* __restrict__ a,
                                             const float4* __restrict__ b, long n4) {
  long i = (long)blockIdx.x * blockDim.x + threadIdx.x;
  if (i < n4) {
    float4 x = a[i];
    float4 y = b[i];
    x.x += y.x; x.y += y.y; x.z += y.z; x.w += y.w;
    a[i] = x;
  }
}

// ---------------------------------------------------------------------------
// SpMM scatter: out[row] += val * x[col], one wave per 32 nonzeros.
// Lanes cooperatively load 32 (row,col,val) triples with NT hints (the index/
// val streams total ~114 MB per call and must not evict the L2-resident dense
// matrices: largest is 35.8 MB vs 192 MB L2). Each triple is broadcast via
// readlane (scalar row/col -> scalar-base addressing); each lane covers 2 of
// the 64 dims with one b64 gather + one dual f32 hardware atomic.
// NOTE: destination accumulates (atomic adds), so scattering into a non-zero
// buffer fuses "acc += spmm(x)".
// ---------------------------------------------------------------------------
__global__ __launch_bounds__(256) void spmm_scatter_k(const int* __restrict__ rows,
                                                      const int* __restrict__ cols,
                                                      const float* __restrict__ vals,
                                                      const float* __restrict__ x,
                                                      float* __restrict__ out,
                                                      int nnz) {
  const int lane = threadIdx.x & 31;
  const int wave = (int)((blockIdx.x * blockDim.x + threadIdx.x) >> 5);
  const int base = wave * 32;
  if (base >= nnz) return;

  int r = 0, c = 0; float v = 0.f;
  const int i = base + lane;
  if (i < nnz) {
    r = __builtin_nontemporal_load(rows + i);
    c = __builtin_nontemporal_load(cols + i);
    v = __builtin_nontemporal_load(vals + i);
  }
  const int cnt = (nnz - base < 32) ? (nnz - base) : 32;

  for (int j = 0; j < cnt; ++j) {
    const int   rr = __shfl(r, j, 32);
    const int   cc = __shfl(c, j, 32);
    const float vv = __shfl(v, j, 32);
    const float2 xv = *(const float2*)(x + (size_t)cc * DIM + lane * 2);
    float* o = out + (size_t)rr * DIM + lane * 2;
    atomic_add2_f32(o, vv * xv.x, vv * xv.y);
  }
}

// ---------------------------------------------------------------------------
// Finalize: accio[row] = accio[row] * 0.25 + prop[row] / max(||prop[row]||, eps)
// One wave handles 16 rows. The 16 squared norms are the diagonal of
// X * X^T (X = 16x64 tile) computed with V_WMMA_F32_16X16X4_F32 accumulated
// over 16 K-chunks. With the ISA's f32 A(16x4)/B(4x16) layouts, the A and B
// register images for the symmetric product are identical:
//   lane L (m = L&15, koff = (L>>4)*2) holds { X[m][4c+koff], X[m][4c+koff+1] }.
// D layout: VGPR r, lanes 0-15 -> D[r][lane]; lanes 16-31 -> D[8+r][lane-16].
// Diagonal D[m][m]: lanes 0-7 have it in c[lane]; lanes 24-31 in c[lane-24].
// Requires nrows % 16 == 0 (holds: 100000 and 40000).
// ---------------------------------------------------------------------------
__global__ __launch_bounds__(256) void finalize_k(const float* __restrict__ prop,
                                                  float* __restrict__ accio,
                                                  int nrows) {
  __shared__ float invn[8 * 16];  // 8 waves per block * 16 rows

  const int lane = threadIdx.x & 31;
  const int wv   = threadIdx.x >> 5;
  const int tile = blockIdx.x * 8 + wv;
  const int row0 = tile * 16;
  if (row0 >= nrows) return;  // wave-uniform exit: EXEC stays all-ones for WMMA

  const float* X = prop + (size_t)row0 * DIM;
  const int m    = lane & 15;
  const int koff = (lane >> 4) * 2;
  const float* xr = X + (size_t)m * DIM + koff;

  v8f c = {};
#pragma unroll
  for (int ch = 0; ch < 16; ++ch) {
    const float2 pr = *(const float2*)(xr + ch * 4);
    v2f a; a.x = pr.x; a.y = pr.y;
    // D = A * A^T + C  (16x16x4 f32)
    c = __builtin_amdgcn_wmma_f32_16x16x4_f32(false, a, false, a,
                                              (short)0, c, false, false);
  }

  // Extract the diagonal element owned by this lane (lanes 0-7 and 24-31).
  const int sel = lane & 7;
  float d = c[0];
  if (sel == 1) d = c[1];
  if (sel == 2) d = c[2];
  if (sel == 3) d = c[3];
  if (sel == 4) d = c[4];
  if (sel == 5) d = c[5];
  if (sel == 6) d = c[6];
  if (sel == 7) d = c[7];
  if (lane < 8 || lane >= 24) {
    const int mrow = (lane < 8) ? lane : (lane - 16);
    invn[wv * 16 + mrow] = 1.0f / fmaxf(sqrtf(d), 1e-12f);
  }
  // Same-wave LDS store->load is in-order (DScnt); no cross-wave sharing.

#pragma unroll 4
  for (int r = 0; r < 16; ++r) {
    const float s = invn[wv * 16 + r];
    const size_t off = (size_t)(row0 + r) * DIM + lane * 2;
    const float2 p = *(const float2*)(prop + off);
    float2* ap = (float2*)(accio + off);
    float2 av = *ap;
    float2 o;
    o.x = av.x * 0.25f + p.x * s;
    o.y = av.y * 0.25f + p.y * s;
    *ap = o;
  }
}

// ---------------------------------------------------------------------------
// Host orchestration: stream-ordered launches enforce layer dependencies.
// d_out layout: [u_ui NU*64 | i_ui NI*64 | ii NI*64 | uu NU*64]
// acc lives in-place in d_out[0 : N*64) (u region + i region are contiguous).
// ---------------------------------------------------------------------------
extern "C" void kernel_launch(void* const* d_in, const int* in_sizes, int n_in,
                              void* d_out, int out_size, void* d_ws, size_t ws_size,
                              hipStream_t stream) {
  const float* user_ui = (const float*)d_in[0];
  const float* item_ui = (const float*)d_in[1];
  const float* uu_emb  = (const float*)d_in[2];
  const float* ii_emb  = (const float*)d_in[3];
  const float* ui_vals = (const float*)d_in[4];
  const float* ii_vals = (const float*)d_in[5];
  const float* uu_vals = (const float*)d_in[6];
  const int*   ui_rows = (const int*)d_in[7];
  const int*   ui_cols = (const int*)d_in[8];
  const int*   ii_rows = (const int*)d_in[9];
  const int*   ii_cols = (const int*)d_in[10];
  const int*   uu_rows = (const int*)d_in[11];
  const int*   uu_cols = (const int*)d_in[12];

  const int NU = in_sizes[0] / DIM;   // 100000
  const int NI = in_sizes[1] / DIM;   // 40000
  const int N  = NU + NI;             // 140000
  const int nnz_ui = in_sizes[4];
  const int nnz_ii = in_sizes[5];
  const int nnz_uu = in_sizes[6];

  float* out    = (float*)d_out;
  float* acc    = out;                              // N*DIM (u region + i region)
  float* out_ii = out + (size_t)N * DIM;            // NI*DIM
  float* out_uu = out_ii + (size_t)NI * DIM;        // NU*DIM

  float* ws    = (float*)d_ws;
  float* ws_ii = ws;                                // NI*DIM
  float* ws_uu = ws_ii + (size_t)NI * DIM;          // NU*DIM
  float* egoA  = ws_uu + (size_t)NU * DIM;          // N*DIM
  float* egoB  = egoA + (size_t)N * DIM;            // N*DIM

  const long ni4 = (long)NI * DIM / 4;
  const long nu4 = (long)NU * DIM / 4;
  const long n4  = (long)N  * DIM / 4;

  const int TB = 256;
  dim3 blk(TB);
  auto blocks4 = [&](long n) { return dim3((unsigned)((n + TB - 1) / TB)); };
  auto blocksS = [&](int nnz) { return dim3((unsigned)((nnz + TB - 1) / TB)); };

  // ---- ii chain: 2 layers ----
  zero_k<<<blocks4(ni4), blk, 0, stream>>>((float4*)ws_ii, ni4);
  spmm_scatter_k<<<blocksS(nnz_ii), blk, 0, stream>>>(ii_rows, ii_cols, ii_vals,
                                                      ii_emb, ws_ii, nnz_ii);
  zero_k<<<blocks4(ni4), blk, 0, stream>>>((float4*)out_ii, ni4);
  spmm_scatter_k<<<blocksS(nnz_ii), blk, 0, stream>>>(ii_rows, ii_cols, ii_vals,
                                                      ws_ii, out_ii, nnz_ii);

  // ---- uu chain: 2 layers ----
  zero_k<<<blocks4(nu4), blk, 0, stream>>>((float4*)ws_uu, nu4);
  spmm_scatter_k<<<blocksS(nnz_uu), blk, 0, stream>>>(uu_rows, uu_cols, uu_vals,
                                                      uu_emb, ws_uu, nnz_uu);
  zero_k<<<blocks4(nu4), blk, 0, stream>>>((float4*)out_uu, nu4);
  spmm_scatter_k<<<blocksS(nnz_uu), blk, 0, stream>>>(uu_rows, uu_cols, uu_vals,
                                                      ws_uu, out_uu, nnz_uu);

  // ---- ui chain: ego0 = concat; acc = ego0; 3 layers of acc += spmm(ego) ----
  concat_k<<<blocks4(n4), blk, 0, stream>>>((const float4*)user_ui,
                                            (const float4*)item_ui,
                                            (float4*)egoA, (float4*)acc, nu4, n4);
  // layer 1: egoA -> egoB (materialized; needed by layer 2), then acc += egoB
  zero_k<<<blocks4(n4), blk, 0, stream>>>((float4*)egoB, n4);
  spmm_scatter_k<<<blocksS(nnz_ui), blk, 0, stream>>>(ui_rows, ui_cols, ui_vals,
                                                      egoA, egoB, nnz_ui);
  add_k<<<blocks4(n4), blk, 0, stream>>>((float4*)acc, (const float4*)egoB, n4);
  // layer 2: egoB -> egoA (materialized; needed by layer 3), then acc += egoA
  zero_k<<<blocks4(n4), blk, 0, stream>>>((float4*)egoA, n4);
  spmm_scatter_k<<<blocksS(nnz_ui), blk, 0, stream>>>(ui_rows, ui_cols, ui_vals,
                                                      egoB, egoA, nnz_ui);
  add_k<<<blocks4(n4), blk, 0, stream>>>((float4*)acc, (const float4*)egoA, n4);
  // layer 3: result only feeds acc -> scatter-accumulate DIRECTLY into acc
  // (saves one full zero pass + one full add pass over N*DIM).
  spmm_scatter_k<<<blocksS(nnz_ui), blk, 0, stream>>>(ui_rows, ui_cols, ui_vals,
                                                      egoA, acc, nnz_ui);

  // ---- finalize: out = acc/4 + l2_normalize(prop), in place over acc ----
  {
    const int tilesU = NU / 16;
    const int tilesI = NI / 16;
    finalize_k<<<dim3((unsigned)((tilesU + 7) / 8)), blk, 0, stream>>>(
        out_uu, acc, NU);                      // users: rows 0..NU-1 of acc
    finalize_k<<<dim3((unsigned)((tilesI + 7) / 8)), blk, 0, stream>>>(
        out_ii, acc + (size_t)NU * DIM, NI);   // items: rows NU..N-1 of acc
  }
}